// MutualExclusivityLoss_11665131176105
// MI455X (gfx1250) — compile-verified
//
#include <hip/hip_runtime.h>
#include <cstdint>

#define T 32
#define ROW_F 36                 // padded LDS row stride in floats (144 B, 16B aligned)
#define TILE_F (32 * ROW_F)      // floats per tile buffer (1152 -> 4608 B)
#define WAVES_PB 6               // 192 threads; 2 buffers/wave keeps LDS = 55,296 B
#define THREADS (WAVES_PB * 32)
#define RTHREADS 256
#define RWAVES 8

// s_final for one row: q_i = A_i * B_i (exclusive prefix/suffix products of 1-p),
// s_i = (s_{i-1} + p_i) * q_i. Two fused passes, ~6 VALU ops/element.
__device__ __forceinline__ float row_loss(const float (&p)[T]) {
  float B[T];
  B[T - 1] = 1.0f;
#pragma unroll
  for (int i = T - 2; i >= 0; --i) B[i] = B[i + 1] * (1.0f - p[i + 1]);
  float A = 1.0f, s = 0.0f;
#pragma unroll
  for (int i = 0; i < T; ++i) {
    float q = A * B[i];
    s = (s + p[i]) * q;
    A *= (1.0f - p[i]);
  }
  return s;
}

__global__ __launch_bounds__(THREADS) void mex_main(const float* __restrict__ probs,
                                                    float* __restrict__ partials,
                                                    int n_rows) {
  // Per wave: 2 tile buffers (double-buffered async pipeline).
  __shared__ __align__(16) float smemf[WAVES_PB * 2 * TILE_F];
  __shared__ float redsh[WAVES_PB];

  const int lane = threadIdx.x & 31;
  const int wave = threadIdx.x >> 5;
  const int gwave = blockIdx.x * WAVES_PB + wave;
  const int nwaves = gridDim.x * WAVES_PB;
  const int fullTiles = n_rows >> 5;   // 32 rows per tile (one row per lane)

  const uint32_t ldsBase = (uint32_t)(uintptr_t)(void*)&smemf[0];
  const uint32_t ldsWaveByte = ldsBase + (uint32_t)(wave * 2 * TILE_F) * 4u;
  // Per-lane sub-address inside a 4KB tile transfer: instruction i, lane L moves
  // global bytes [i*512 + L*16, +16) -> LDS row (i*4 + L/8) at col (L%8)*16.
  const uint32_t ldsLaneByte =
      (uint32_t)((lane >> 3) * (ROW_F * 4) + (lane & 7) * 16);

  float sum = 0.0f;

  // Issue one tile (8 x async b128, coalesced 512B/instruction, NT streaming).
  auto prefetch = [&](int tileIdx, int b) {
    unsigned long long g = (unsigned long long)(uintptr_t)probs +
                           (unsigned long long)tileIdx * 4096ull +
                           (unsigned long long)(lane * 16);
    uint32_t l = ldsWaveByte + (uint32_t)(b * TILE_F) * 4u + ldsLaneByte;
#pragma unroll
    for (int i = 0; i < 8; ++i) {
      asm volatile("global_load_async_to_lds_b128 %0, %1, off th:TH_LOAD_NT"
                   :: "v"(l), "v"(g) : "memory");
      l += 4u * ROW_F * 4u;   // +4 LDS rows (576 B, keeps 16B alignment)
      g += 512ull;            // +4 memory rows
    }
  };

  int tile = gwave;
  int buf = 0;
  if (tile < fullTiles) prefetch(tile, 0);
  while (tile < fullTiles) {
    const int next = tile + nwaves;
    if (next < fullTiles) {
      // WAR safety: reads of buf^1 (consumed last iteration) must be complete.
      asm volatile("s_wait_dscnt 0" ::: "memory");
      prefetch(next, buf ^ 1);
      // Async loads complete in order: <=8 outstanding means current buffer landed.
      asm volatile("s_wait_asynccnt 0x8" ::: "memory");
    } else {
      asm volatile("s_wait_asynccnt 0x0" ::: "memory");
    }

    // Each lane consumes its own padded row from LDS (8 x ds_load_b128).
    const int rowF = wave * 2 * TILE_F + buf * TILE_F + lane * ROW_F;
    float p[T];
#pragma unroll
    for (int k = 0; k < 8; ++k) {
      float4 v = *reinterpret_cast<const float4*>(&smemf[rowF + 4 * k]);
      p[4 * k + 0] = v.x; p[4 * k + 1] = v.y;
      p[4 * k + 2] = v.z; p[4 * k + 3] = v.w;
    }
    sum += row_loss(p);

    buf ^= 1;
    tile = next;
  }

  // Tail rows (n_rows % 32 != 0): direct global float4 loads, one row per thread.
  for (int r = (fullTiles << 5) + blockIdx.x * THREADS + threadIdx.x;
       r < n_rows; r += gridDim.x * THREADS) {
    const float4* rp = reinterpret_cast<const float4*>(probs) + (size_t)r * 8;
    float p[T];
#pragma unroll
    for (int k = 0; k < 8; ++k) {
      float4 v = rp[k];
      p[4 * k + 0] = v.x; p[4 * k + 1] = v.y;
      p[4 * k + 2] = v.z; p[4 * k + 3] = v.w;
    }
    sum += row_loss(p);
  }

  // wave32 reduce, then block reduce to one partial per block (deterministic).
#pragma unroll
  for (int o = 16; o > 0; o >>= 1) sum += __shfl_xor(sum, o, 32);
  if (lane == 0) redsh[wave] = sum;
  __syncthreads();
  if (threadIdx.x == 0) {
    float s = 0.0f;
#pragma unroll
    for (int i = 0; i < WAVES_PB; ++i) s += redsh[i];
    partials[blockIdx.x] = s;
  }
}

__global__ __launch_bounds__(RTHREADS) void mex_reduce(const float* __restrict__ partials,
                                                       int n, float* __restrict__ out,
                                                       float invN) {
  __shared__ float sh[RWAVES];
  float s = 0.0f;
  for (int i = threadIdx.x; i < n; i += RTHREADS) s += partials[i];
#pragma unroll
  for (int o = 16; o > 0; o >>= 1) s += __shfl_xor(s, o, 32);
  const int lane = threadIdx.x & 31;
  const int wave = threadIdx.x >> 5;
  if (lane == 0) sh[wave] = s;
  __syncthreads();
  if (threadIdx.x == 0) {
    float t = 0.0f;
#pragma unroll
    for (int i = 0; i < RWAVES; ++i) t += sh[i];
    out[0] = -t * invN;
  }
}

extern "C" void kernel_launch(void* const* d_in, const int* in_sizes, int n_in,
                              void* d_out, int out_size, void* d_ws, size_t ws_size,
                              hipStream_t stream) {
  const float* probs = (const float*)d_in[0];
  const long long total = (long long)in_sizes[0];   // N * 32
  const int n_rows = (int)(total / T);

  int blocks = 1536;   // 9216 waves -> ~6.8 tiles/wave, deep async pipeline
  const int maxb = (int)(ws_size / sizeof(float));
  if (maxb > 0 && maxb < blocks) blocks = maxb;

  float* partials = (float*)d_ws;
  mex_main<<<blocks, THREADS, 0, stream>>>(probs, partials, n_rows);
  mex_reduce<<<1, RTHREADS, 0, stream>>>(partials, blocks, (float*)d_out,
                                         1.0f / (float)n_rows);
}